// MSLCrossEntropyLoss_4_57037165690977
// MI455X (gfx1250) — compile-verified
//
#include <hip/hip_runtime.h>
#include <stdint.h>

// ---------------- problem constants (fixed by the reference) ----------------
#define NCLS 1000              // C
#define TAU  1e-5f
#define RPT  8                 // rows per TDM tile (8 * 1000 * 4 = 32 KB, x2 buffers = 64 KB LDS)
#define LOG2E 1.4426950408889634f

typedef unsigned int uint32x4 __attribute__((ext_vector_type(4)));
typedef int          int32x8  __attribute__((ext_vector_type(8)));
typedef int          int32x4  __attribute__((ext_vector_type(4)));
typedef float        v2f      __attribute__((ext_vector_type(2)));
typedef float        v8f      __attribute__((ext_vector_type(8)));

// ---------------------------------------------------------------------------
// TDM: issue an async load of `rows` rows of NCLS f32 (row stride NCLS) from
// global into LDS.  Descriptor packed per CDNA5 ISA ch.8 (D# groups 0/1).
// 2D tile -> higher-dim groups zero.  Tracked with TENSORcnt.
// This toolchain exposes the 6-arg builtin:
//   (uint32x4 g0, int32x8 g1, int32x4 g2, int32x4 g3, int32x8 g4, i32 cpol)
// ---------------------------------------------------------------------------
__device__ __forceinline__ void tdm_load_rows(const float* gsrc, void* lds_dst,
                                              int rows, int rows_avail) {
  uint64_t ga = (uint64_t)(uintptr_t)gsrc;               // 57-bit global byte addr
  uint32_t la = (uint32_t)(uintptr_t)lds_dst;            // LDS aperture: low 32 bits = LDS byte offset
  uint32x4 g0;
  g0.x = 1u;                                             // count=1 valid descriptor
  g0.y = la;                                             // lds_addr
  g0.z = (uint32_t)ga;                                   // global_addr[31:0]
  g0.w = (uint32_t)((ga >> 32) & 0x01FFFFFFu) | (2u << 30); // global_addr[56:32] | type=2
  uint32_t dim1 = (uint32_t)rows_avail;
  int32x8 g1;
  g1[0] = (int)(2u << 16);                               // workgroup_mask=0, data_size=2 (4B)
  g1[1] = (int)(((uint32_t)NCLS & 0xFFFFu) << 16);       // tensor_dim0[15:0] @ bits 63:48
  g1[2] = (int)(((uint32_t)NCLS >> 16) | ((dim1 & 0xFFFFu) << 16)); // dim0 hi | tensor_dim1 lo
  g1[3] = (int)((dim1 >> 16) | ((uint32_t)NCLS << 16));  // tensor_dim1 hi | tile_dim0=NCLS
  g1[4] = (int)((uint32_t)rows & 0xFFFFu);               // tile_dim1=rows, tile_dim2=0
  g1[5] = (int)(uint32_t)NCLS;                           // tensor_dim0_stride[31:0]
  g1[6] = 0;                                             // stride0 hi | stride1 lo
  g1[7] = 0;
  int32x4 g2 = {0, 0, 0, 0};
  int32x4 g3 = {0, 0, 0, 0};
  int32x8 g4 = {0, 0, 0, 0, 0, 0, 0, 0};
  __builtin_amdgcn_tensor_load_to_lds(g0, g1, g2, g3, g4, 0);
}

// ---------------------------------------------------------------------------
// Kernel 0: zero the histogram + accumulators each call (graph replays).
// ---------------------------------------------------------------------------
__global__ void k_zero(unsigned* hist, float* accum) {
  int i = threadIdx.x;
  if (i < 1024) hist[i] = 0u;
  if (i < 4)    accum[i] = 0.0f;
}

// ---------------------------------------------------------------------------
// Kernel 1: per-row softmax stats via TDM double-buffered LDS tiles.
// Workgroup = 256 threads = 8 waves; each wave owns one row of an 8-row tile.
// Produces: pvals[row] = softmax prob of target class; hist[pred]++.
// ---------------------------------------------------------------------------
__global__ void __launch_bounds__(256)
k_rowstats(const float* __restrict__ logits, const int* __restrict__ targets,
           float* __restrict__ pvals, unsigned* __restrict__ hist,
           int nrows, int rows_per_wg) {
  __shared__ float tiles[2][RPT * NCLS];

  const int row0   = blockIdx.x * rows_per_wg;
  const int wave   = threadIdx.x >> 5;
  const int lane   = threadIdx.x & 31;
  const int ntiles = rows_per_wg / RPT;

  // prime the pipeline: tile 0 -> buffer 0
  if (threadIdx.x == 0)
    tdm_load_rows(logits + (size_t)row0 * NCLS, &tiles[0][0], RPT,
                  nrows - row0 > 0 ? nrows - row0 : 1);

  for (int t = 0; t < ntiles; ++t) {
    const int cur = t & 1;
    if (threadIdx.x == 0) {
      if (t + 1 < ntiles) {
        int r1 = row0 + (t + 1) * RPT;
        tdm_load_rows(logits + (size_t)r1 * NCLS, &tiles[cur ^ 1][0], RPT,
                      nrows - r1 > 0 ? nrows - r1 : 1);
        __builtin_amdgcn_s_wait_tensorcnt(1);   // tile t complete, t+1 in flight
      } else {
        __builtin_amdgcn_s_wait_tensorcnt(0);
      }
    }
    __syncthreads();                             // tile t visible to all waves

    const int row = row0 + t * RPT + wave;
    const float* r = &tiles[cur][wave * NCLS];

    // pass 1 (LDS): max + argmax (first-occurrence tie-break, like jnp.argmax)
    float m = -3.402823466e+38f;
    int   mi = NCLS;
    for (int k = lane; k < NCLS; k += 32) {
      float x = r[k];
      if (x > m || (x == m && k < mi)) { m = x; mi = k; }
    }
    for (int off = 16; off > 0; off >>= 1) {
      float om = __shfl_xor(m, off, 32);
      int   oi = __shfl_xor(mi, off, 32);
      if (om > m || (om == m && oi < mi)) { m = om; mi = oi; }
    }

    // pass 2 (LDS): sum of exp(x - m)  (v_exp_f32 on the TRANS pipe)
    float s = 0.0f;
    for (int k = lane; k < NCLS; k += 32)
      s += exp2f((r[k] - m) * LOG2E);
    for (int off = 16; off > 0; off >>= 1) s += __shfl_xor(s, off, 32);

    if (lane == 0 && row < nrows) {
      int tgt = targets[row];
      float pt = exp2f((r[tgt] - m) * LOG2E) / s;
      pvals[row] = pt;
      atomicAdd(&hist[mi], 1u);
    }
    __syncthreads();                             // done reading buffer before TDM reuses it
  }
}

// ---------------------------------------------------------------------------
// Kernel 2: col_sum[j] = ones^T * conf_N (:,j) + hist[j]  via WMMA f32 16x16x4,
// then bias[j] = TAU * col_sum^-(1/4) and bias_sum accumulated.
// One wave covers 16 columns: A = ones(16x4), B = conf 4x16 tile, 250 MACs.
// ---------------------------------------------------------------------------
__global__ void __launch_bounds__(256)
k_bias(const float* __restrict__ conf, const unsigned* __restrict__ hist,
       float* __restrict__ bias, float* __restrict__ accum) {
  const int wid  = (blockIdx.x * blockDim.x + threadIdx.x) >> 5;
  const int lane = threadIdx.x & 31;
  const int j0   = wid * 16;
  if (j0 >= NCLS) return;                        // wave-uniform: EXEC stays all-1s for WMMA

  const int n  = lane & 15;
  const int kh = lane >> 4;                      // which K-half this lane holds
  const int j  = j0 + n;
  const int jj = (j < NCLS) ? j : (NCLS - 1);    // clamp tail loads (results discarded)

  v2f a; a.x = 1.0f; a.y = 1.0f;                 // A = ones(16x4)
  v8f c = {0.f, 0.f, 0.f, 0.f, 0.f, 0.f, 0.f, 0.f};
  for (int k0 = 0; k0 < NCLS; k0 += 4) {
    v2f b;
    b.x = conf[(size_t)(k0 + 2 * kh)     * NCLS + jj];   // B[K=0/2][n]
    b.y = conf[(size_t)(k0 + 2 * kh + 1) * NCLS + jj];   // B[K=1/3][n]
    c = __builtin_amdgcn_wmma_f32_16x16x4_f32(false, a, false, b,
                                              (short)0, c, false, false);
  }
  // D[m][n] = colsum(j0+n) for every m; take row M=0 -> c[0] on lanes 0..15
  if (lane < 16 && j < NCLS) {
    float cs = c[0] + (float)hist[j];
    cs = (cs == 0.0f) ? 1e-8f : cs;
    float b_ = TAU * rsqrtf(sqrtf(cs));          // cs^(-0.25)
    bias[j] = b_;
    atomicAdd(&accum[1], b_);                    // bias_sum
  }
}

// ---------------------------------------------------------------------------
// Kernel 3: per-row loss body  -log(p_t - bias[t]),  wave+block reduce, atomic.
// ---------------------------------------------------------------------------
__global__ void __launch_bounds__(256)
k_loss(const float* __restrict__ pvals, const int* __restrict__ targets,
       const float* __restrict__ bias, float* __restrict__ accum, int nrows) {
  int i = blockIdx.x * blockDim.x + threadIdx.x;
  float v = 0.0f;
  if (i < nrows) {
    int t = targets[i];
    v = -__logf(pvals[i] - bias[t]);
  }
  for (int off = 16; off > 0; off >>= 1) v += __shfl_xor(v, off, 32);
  __shared__ float wsum[8];
  int lane = threadIdx.x & 31, wv = threadIdx.x >> 5;
  if (lane == 0) wsum[wv] = v;
  __syncthreads();
  if (threadIdx.x == 0) {
    float s = 0.0f;
    #pragma unroll
    for (int w = 0; w < 8; ++w) s += wsum[w];
    atomicAdd(&accum[0], s);
  }
}

// ---------------------------------------------------------------------------
// Kernel 4: mean = (sum_i -log(y_i) + N*log(1 - bias_sum)) / N
// (row prob-sum is 1, so denom = 1 - sum_j bias[j], row-constant)
// ---------------------------------------------------------------------------
__global__ void k_final(const float* __restrict__ accum, float* __restrict__ out,
                        int nrows) {
  if (threadIdx.x == 0 && blockIdx.x == 0) {
    float denom = 1.0f - accum[1];
    out[0] = (accum[0] + (float)nrows * __logf(denom)) / (float)nrows;
  }
}

// ---------------------------------------------------------------------------
extern "C" void kernel_launch(void* const* d_in, const int* in_sizes, int n_in,
                              void* d_out, int out_size, void* d_ws, size_t ws_size,
                              hipStream_t stream) {
  const float* logits  = (const float*)d_in[0];   // [N, 1000] f32
  const int*   targets = (const int*)  d_in[1];   // [N] int
  const float* conf    = (const float*)d_in[2];   // [1000, 1000] f32
  const int nrows = in_sizes[1];

  // workspace layout
  char* ws = (char*)d_ws;
  float*    pvals = (float*)ws;                                  // N floats
  unsigned* hist  = (unsigned*)(ws + (size_t)nrows * 4);         // 1024 u32
  float*    bias  = (float*)(ws + (size_t)nrows * 4 + 4096);     // 1024 f32
  float*    accum = (float*)(ws + (size_t)nrows * 4 + 8192);     // [loss_sum, bias_sum, ...]

  k_zero<<<1, 1024, 0, stream>>>(hist, accum);

  const int rows_per_wg = 64;                    // 8 tiles of 8 rows per workgroup
  const int blocks = (nrows + rows_per_wg - 1) / rows_per_wg;
  k_rowstats<<<blocks, 256, 0, stream>>>(logits, targets, pvals, hist,
                                         nrows, rows_per_wg);

  k_bias<<<8, 256, 0, stream>>>(conf, hist, bias, accum);   // 64 waves, 63 used

  k_loss<<<(nrows + 255) / 256, 256, 0, stream>>>(pvals, targets, bias, accum, nrows);

  k_final<<<1, 32, 0, stream>>>(accum, (float*)d_out, nrows);
}